// DCNv4_1D_CUDA_27513560498562
// MI455X (gfx1250) — compile-verified
//
#include <hip/hip_runtime.h>
#include <math.h>

// DCNv4-1D: LayerNorm + GELU + offset/mask projection (fp32 WMMA) + deformable sampling
// N=8, C=256, L=8192, K=3, G=8, GC=32, O=2*G*K=48

#define N_ 8
#define C_ 256
#define L_ 8192
#define K_ 3
#define G_ 8
#define GC_ 32
#define O_ 48
#define TILE 16
#define XS_STRIDE 257   // 256 + 1 pad: bank-conflict-free for both l-major and c-major access

typedef __attribute__((ext_vector_type(2))) float v2f;
typedef __attribute__((ext_vector_type(8))) float v8f;

__device__ __forceinline__ float gelu_exact(float x) {
    // 0.5 * x * (1 + erf(x / sqrt(2))) -- matches jax.nn.gelu(approximate=False)
    return 0.5f * x * (1.0f + erff(x * 0.70710678118654752440f));
}

// ---------------------------------------------------------------------------
// Pass 1: per 16-row (l) tile: LayerNorm over C, GELU, store xa in [N,C,L]
// layout, then project to om[.,48] with V_WMMA_F32_16X16X4_F32.
// block = 128 threads (4 wave32), grid = N*L/16
// ---------------------------------------------------------------------------
__global__ __launch_bounds__(128) void dcn_ln_gelu_proj(
    const float* __restrict__ x,      // [N,C,L]
    const float* __restrict__ ln_w,   // [C]
    const float* __restrict__ ln_b,   // [C]
    const float* __restrict__ om_w,   // [O,C] row-major
    const float* __restrict__ om_b,   // [O]
    float* __restrict__ xa_ws,        // [N,C,L]
    float* __restrict__ om_ws)        // [N*L, O]
{
    __shared__ float xs[TILE * XS_STRIDE];
    __shared__ float mu_s[TILE];
    __shared__ float rstd_s[TILE];

    const int tid  = threadIdx.x;
    const int row0 = blockIdx.x * TILE;       // flattened (n*L + l) row
    const int n    = row0 / L_;
    const int l0   = row0 - n * L_;

    // ---- load x tile: xs[li][c], coalesced along l (16 consecutive floats) ----
    const float* xbase = x + (size_t)n * C_ * L_ + l0;
    #pragma unroll
    for (int i = 0; i < (TILE * C_) / 128; ++i) {
        int idx = tid + i * 128;
        int li  = idx & (TILE - 1);
        int c   = idx >> 4;
        xs[li * XS_STRIDE + c] = xbase[(size_t)c * L_ + li];
    }
    __syncthreads();

    // ---- LayerNorm stats: 8 lanes per l-row, butterfly reduce within wave ----
    {
        int li = tid >> 3;
        int j  = tid & 7;
        float s = 0.f, sq = 0.f;
        #pragma unroll
        for (int c = 0; c < C_; c += 8) {
            float v = xs[li * XS_STRIDE + c + j];
            s += v; sq += v * v;
        }
        #pragma unroll
        for (int m = 1; m < 8; m <<= 1) {
            s  += __shfl_xor(s,  m, 8);
            sq += __shfl_xor(sq, m, 8);
        }
        if (j == 0) {
            float mu  = s * (1.0f / C_);
            float var = sq * (1.0f / C_) - mu * mu;
            mu_s[li]   = mu;
            rstd_s[li] = rsqrtf(var + 1e-6f);
        }
    }
    __syncthreads();

    // ---- normalize + affine + GELU; write xa to LDS and to [N,C,L] global ----
    float* xa_base = xa_ws + (size_t)n * C_ * L_ + l0;
    #pragma unroll
    for (int i = 0; i < (TILE * C_) / 128; ++i) {
        int idx = tid + i * 128;
        int li  = idx & (TILE - 1);
        int c   = idx >> 4;
        float v  = xs[li * XS_STRIDE + c];
        float xn = (v - mu_s[li]) * rstd_s[li] * ln_w[c] + ln_b[c];
        float xa = gelu_exact(xn);
        xs[li * XS_STRIDE + c] = xa;
        xa_base[(size_t)c * L_ + li] = xa;     // coalesced: 16 consecutive l
    }
    __syncthreads();

    // ---- projection: om[16, 48] = xa_tile[16, 256] x om_w^T[256, 48] ----
    // Waves 0..2 each own one 16-wide output-column tile; full fp32 WMMA.
    const int wave = tid >> 5;
    if (wave < 3) {
        const int lane = tid & 31;
        const int half = lane >> 4;        // 0: lanes 0-15, 1: lanes 16-31
        const int lid  = lane & 15;
        const int koff = half * 2;         // ISA A/B layout: VGPR0=K{0,2}, VGPR1=K{1,3}
        const int n0   = wave * 16;        // output column tile base

        const float* arow = xs + lid * XS_STRIDE;            // A: row M=lid
        const float* wrow = om_w + (size_t)(n0 + lid) * C_;  // B: col N=lid -> om_w row

        v8f acc = {};
        #pragma unroll 4
        for (int k = 0; k < C_; k += 4) {
            v2f a, b;
            a.x = arow[k + koff];
            a.y = arow[k + koff + 1];
            b.x = wrow[k + koff];
            b.y = wrow[k + koff + 1];
            acc = __builtin_amdgcn_wmma_f32_16x16x4_f32(false, a, false, b,
                                                        (short)0, acc, false, false);
        }

        // D layout: VGPR r -> (M = r + 8*half, N = lid); add bias, store
        float bias = om_b[n0 + lid];
        #pragma unroll
        for (int r = 0; r < 8; ++r) {
            int m = r + half * 8;
            om_ws[(size_t)(row0 + m) * O_ + (n0 + lid)] = acc[r] + bias;
        }
    }
}

// ---------------------------------------------------------------------------
// Pass 2: deformable bilinear sampling + mask-weighted aggregation.
// One thread per (n, g, l); loops over the 32 group channels.
// block = 256 threads, grid = N * G * (L/256)
// ---------------------------------------------------------------------------
__global__ __launch_bounds__(256) void dcn_sample(
    const float* __restrict__ xa_ws,  // [N,C,L]
    const float* __restrict__ om_ws,  // [N*L, O]
    float* __restrict__ out)          // [N,C,L]
{
    const int t   = threadIdx.x;
    int bid = blockIdx.x;
    const int lt = bid & 31;  bid >>= 5;   // L/256 = 32 tiles
    const int g  = bid & 7;   bid >>= 3;
    const int n  = bid;
    const int l  = lt * 256 + t;

    const float* omr = om_ws + (size_t)(n * L_ + l) * O_ + g * K_;
    float offs[K_] = { omr[0], omr[1], omr[2] };
    float mk[K_]   = { omr[G_ * K_ + 0], omr[G_ * K_ + 1], omr[G_ * K_ + 2] };

    int   i0[K_], i1[K_];
    float w0[K_], w1[K_];
    #pragma unroll
    for (int k = 0; k < K_; ++k) {
        float p  = (float)(l - 1 + k) + offs[k];    // base grid (pad=1, dil=1) + offset
        float p0 = floorf(p);
        float f1 = p - p0;
        float f0 = 1.0f - f1;
        int a = (int)p0;
        int b = a + 1;
        w0[k] = ((a >= 0) && (a < L_)) ? f0 * mk[k] : 0.0f;
        w1[k] = ((b >= 0) && (b < L_)) ? f1 * mk[k] : 0.0f;
        i0[k] = min(max(a, 0), L_ - 1);
        i1[k] = min(max(b, 0), L_ - 1);
    }

    const float* xp = xa_ws + ((size_t)n * C_ + (size_t)g * GC_) * L_;
    float*       op = out   + ((size_t)n * C_ + (size_t)g * GC_) * L_ + l;

    #pragma unroll 4
    for (int c = 0; c < GC_; ++c) {
        const float* xc = xp + (size_t)c * L_;
        float s = 0.0f;
        #pragma unroll
        for (int k = 0; k < K_; ++k)
            s += w0[k] * xc[i0[k]] + w1[k] * xc[i1[k]];
        op[(size_t)c * L_] = s;                     // coalesced along l
    }
}

// ---------------------------------------------------------------------------
extern "C" void kernel_launch(void* const* d_in, const int* in_sizes, int n_in,
                              void* d_out, int out_size, void* d_ws, size_t ws_size,
                              hipStream_t stream) {
    const float* x    = (const float*)d_in[0];
    const float* ln_w = (const float*)d_in[1];
    const float* ln_b = (const float*)d_in[2];
    const float* om_w = (const float*)d_in[3];
    const float* om_b = (const float*)d_in[4];
    float* out = (float*)d_out;

    float* xa_ws = (float*)d_ws;                        // N*C*L floats = 64 MB
    float* om_ws = xa_ws + (size_t)N_ * C_ * L_;        // N*L*48 floats = 12.6 MB

    dcn_ln_gelu_proj<<<(N_ * L_) / TILE, 128, 0, stream>>>(
        x, ln_w, ln_b, om_w, om_b, xa_ws, om_ws);

    dcn_sample<<<N_ * G_ * (L_ / 256), 256, 0, stream>>>(
        xa_ws, om_ws, out);
}